// GraphAttentionLayer_61160334295334
// MI455X (gfx1250) — compile-verified
//
#include <hip/hip_runtime.h>
#include <math.h>

#define N_NODES 8192
#define DIM     256
#define ALPHA   0.2f

typedef __attribute__((ext_vector_type(2))) float v2f;
typedef __attribute__((ext_vector_type(8))) float v8f;

// ---------------------------------------------------------------------------
// Kernel A: Wh = h @ W using V_WMMA_F32_16X16X4_F32.
// One wave (32 lanes) computes one 16x16 tile of Wh; K swept in steps of 4.
// A 16x4 layout (ISA 7.12.2): lanes 0-15 hold M=lane, VGPR0=K0,VGPR1=K1;
// lanes 16-31 hold M=lane-16, K2/K3.  B 4x16 mirrors with N across lanes.
// D: VGPR v -> (M=v, N=lane) for lanes 0-15, (M=v+8, N=lane-16) for 16-31.
// ---------------------------------------------------------------------------
__global__ __launch_bounds__(256) void gat_gemm_wh(const float* __restrict__ h,
                                                   const float* __restrict__ W,
                                                   float* __restrict__ Wh) {
    const int lane = threadIdx.x & 31;
    const int wave = threadIdx.x >> 5;
    const int tile = blockIdx.x * 8 + wave;      // 8192 tiles total
    const int m0 = (tile >> 4) * 16;             // 512 M-tiles
    const int n0 = (tile & 15) * 16;             // 16  N-tiles
    const int l15 = lane & 15;
    const int kh  = (lane >> 4) * 2;             // 0 or 2

    const float* arow = h + (size_t)(m0 + l15) * DIM + kh;  // A[M=l15, K=kh..kh+1]
    const float* bcol = W + (size_t)kh * DIM + (n0 + l15);  // B[K=kh..kh+1, N=l15]

    v8f c = {};
#pragma unroll 4
    for (int k = 0; k < DIM; k += 4) {
        v2f a;
        a.x = arow[k];
        a.y = arow[k + 1];
        v2f b;
        b.x = bcol[(size_t)k * DIM];
        b.y = bcol[(size_t)(k + 1) * DIM];
        c = __builtin_amdgcn_wmma_f32_16x16x4_f32(false, a, false, b,
                                                  (short)0, c, false, false);
    }

    const int mo = (lane >> 4) * 8;
    float* out = Wh + (size_t)(m0 + mo) * DIM + n0 + l15;
#pragma unroll
    for (int v = 0; v < 8; ++v) out[(size_t)v * DIM] = c[v];
}

// ---------------------------------------------------------------------------
// Kernel B: s1[i] = Wh[i,:]·a[:256],  s2[i] = Wh[i,:]·a[256:].
// One wave per row; 8 features per lane; shfl_xor tree reduce (wave32).
// ---------------------------------------------------------------------------
__global__ __launch_bounds__(256) void gat_scores(const float* __restrict__ Wh,
                                                  const float* __restrict__ a,
                                                  float* __restrict__ s1,
                                                  float* __restrict__ s2) {
    const int lane = threadIdx.x & 31;
    const int wave = threadIdx.x >> 5;
    const int row  = blockIdx.x * 8 + wave;
    const float* wr = Wh + (size_t)row * DIM;

    float d1 = 0.f, d2 = 0.f;
#pragma unroll
    for (int q = 0; q < 8; ++q) {
        const int f = lane + q * 32;
        const float v = wr[f];
        d1 = fmaf(v, a[f], d1);
        d2 = fmaf(v, a[DIM + f], d2);
    }
#pragma unroll
    for (int off = 16; off > 0; off >>= 1) {
        d1 += __shfl_xor(d1, off, 32);
        d2 += __shfl_xor(d2, off, 32);
    }
    if (lane == 0) { s1[row] = d1; s2[row] = d2; }
}

// ---------------------------------------------------------------------------
// Kernel B2: global max of s2 -> softmax stability bound (shift-invariant).
// ---------------------------------------------------------------------------
__global__ __launch_bounds__(256) void gat_maxs2(const float* __restrict__ s2,
                                                 float* __restrict__ mx) {
    __shared__ float red[256];
    float m = -INFINITY;
    for (int i = threadIdx.x; i < N_NODES; i += 256) m = fmaxf(m, s2[i]);
    red[threadIdx.x] = m;
    __syncthreads();
    for (int s = 128; s > 0; s >>= 1) {
        if (threadIdx.x < s) red[threadIdx.x] = fmaxf(red[threadIdx.x], red[threadIdx.x + s]);
        __syncthreads();
    }
    if (threadIdx.x == 0) mx[0] = red[0];
}

// ---------------------------------------------------------------------------
// Kernel C: single pass over adj row i.  p_j = exp(lrelu(s1_i+s2_j) - bound)
// for neighbors; compact (j,p) into LDS; sparse-gather Wh rows (thread=feat);
// normalize by block-reduced sum; ELU; store.  adj read exactly once (256MB).
// ---------------------------------------------------------------------------
__global__ __launch_bounds__(256) void gat_aggregate(const float* __restrict__ adj,
                                                     const float* __restrict__ Wh,
                                                     const float* __restrict__ s1,
                                                     const float* __restrict__ s2,
                                                     const float* __restrict__ mxs2,
                                                     float* __restrict__ out) {
    __shared__ int   s_j[1024];
    __shared__ float s_p[1024];
    __shared__ int   s_cnt;
    __shared__ float red[256];

    const int i   = blockIdx.x;
    const int tid = threadIdx.x;

    const float si = s1[i];
    float bound = si + mxs2[0];
    bound = bound > 0.f ? bound : ALPHA * bound;   // lrelu(upper bound) >= all e[i,j]

    if (tid == 0) s_cnt = 0;
    __syncthreads();

    const float* adjRow = adj + (size_t)i * N_NODES;
    float acc  = 0.f;   // h_prime[i, tid]
    float lsum = 0.f;   // softmax denominator (partial)

    for (int j0 = 0; j0 < N_NODES; j0 += 1024) {
        if (j0 + 1024 < N_NODES)
            __builtin_prefetch(adjRow + j0 + 1024 + tid * 4, 0, 0);

        const float4 a4 = *(const float4*)(adjRow + j0 + tid * 4);
#pragma unroll
        for (int q = 0; q < 4; ++q) {
            const float av = (&a4.x)[q];
            if (av > 0.f) {
                const int j = j0 + tid * 4 + q;
                float e = si + s2[j];
                e = e > 0.f ? e : ALPHA * e;
                const float p = __expf(e - bound);
                const int idx = atomicAdd(&s_cnt, 1);
                s_j[idx] = j;
                s_p[idx] = p;
                lsum += p;
            }
        }
        __syncthreads();
        const int cnt = s_cnt;
        for (int k = 0; k < cnt; ++k)
            acc = fmaf(s_p[k], Wh[(size_t)s_j[k] * DIM + tid], acc);
        __syncthreads();
        if (tid == 0) s_cnt = 0;
        __syncthreads();
    }

    red[tid] = lsum;
    __syncthreads();
    for (int s = 128; s > 0; s >>= 1) {
        if (tid < s) red[tid] += red[tid + s];
        __syncthreads();
    }
    const float l  = red[0];
    const float hp = acc / l;
    out[(size_t)i * DIM + tid] = hp > 0.f ? hp : (__expf(hp) - 1.f);
}

// ---------------------------------------------------------------------------
extern "C" void kernel_launch(void* const* d_in, const int* in_sizes, int n_in,
                              void* d_out, int out_size, void* d_ws, size_t ws_size,
                              hipStream_t stream) {
    const float* h   = (const float*)d_in[0];   // [8192, 256]
    const float* adj = (const float*)d_in[1];   // [8192, 8192]
    const float* W   = (const float*)d_in[2];   // [256, 256]
    const float* a   = (const float*)d_in[3];   // [512, 1]
    float* out = (float*)d_out;                 // [8192, 256]

    float* Wh = (float*)d_ws;                   // 8192*256 f32 = 8 MB
    float* s1 = Wh + (size_t)N_NODES * DIM;     // 8192 f32
    float* s2 = s1 + N_NODES;                   // 8192 f32
    float* mx = s2 + N_NODES;                   // 1 f32

    (void)in_sizes; (void)n_in; (void)out_size; (void)ws_size;

    // A: 8192 16x16 tiles, 8 waves/block -> 1024 blocks
    gat_gemm_wh<<<1024, 256, 0, stream>>>(h, W, Wh);
    // B: 8 rows/block -> 1024 blocks
    gat_scores<<<1024, 256, 0, stream>>>(Wh, a, s1, s2);
    gat_maxs2<<<1, 256, 0, stream>>>(s2, mx);
    // C: one block per graph node
    gat_aggregate<<<N_NODES, 256, 0, stream>>>(adj, Wh, s1, s2, mx, out);
}